// Cafi_model_67113158967337
// MI455X (gfx1250) — compile-verified
//
#include <hip/hip_runtime.h>
#include <hip/hip_bf16.h>
#include <math.h>

typedef __attribute__((ext_vector_type(2))) float v2f;
typedef __attribute__((ext_vector_type(8))) float v8f;

#define B_ 8
#define S_ 64
#define V_ 32768
#define QUAD 0.19634954084936207f  /* 4*pi/64 */

// output layout (floats): sampled | E | coords | points_code
#define OFF_E       (B_*V_)                  /* 262144  */
#define OFF_COORDS  (OFF_E + B_*9)           /* 262216  */
#define OFF_PC      (OFF_COORDS + B_*V_*3)   /* 1048648 */

// Solid harmonics: Y_l(x/r)*r^l expressed directly as polynomials in x (r2 = |x|^2).
__device__ __forceinline__ void solid_sh(float X, float Y, float Z, float r2, float* P) {
  float XX = X*X, YY = Y*Y, ZZ = Z*Z;
  P[0]  = 0.28209479f;
  P[1]  = 0.48860251f*Y;
  P[2]  = 0.48860251f*Z;
  P[3]  = 0.48860251f*X;
  P[4]  = 1.09254843f*X*Y;
  P[5]  = 1.09254843f*Y*Z;
  P[6]  = 0.31539157f*(3.f*ZZ - r2);
  P[7]  = 1.09254843f*X*Z;
  P[8]  = 0.54627421f*(XX - YY);
  P[9]  = 0.59004359f*Y*(3.f*XX - YY);
  P[10] = 2.89061144f*X*Y*Z;
  P[11] = 0.45704579f*Y*(5.f*ZZ - r2);
  P[12] = 0.37317633f*Z*(5.f*ZZ - 3.f*r2);
  P[13] = 0.45704579f*X*(5.f*ZZ - r2);
  P[14] = 1.44530572f*Z*(XX - YY);
  P[15] = 0.59004359f*X*(XX - 3.f*YY);
}

// ---------------------------------------------------------------------------
// Stage 1: per-batch MLPs -> coeffs C[16][64] (to workspace) and frame E (to out)
// ---------------------------------------------------------------------------
__global__ __launch_bounds__(256) void stage1_kernel(
    const float* __restrict__ F,  const float* __restrict__ S2,
    const float* __restrict__ bw1, const float* __restrict__ bb1,
    const float* __restrict__ bw2, const float* __restrict__ bb2,
    const float* __restrict__ cw1, const float* __restrict__ cb1,
    const float* __restrict__ cw2, const float* __restrict__ cb2,
    float* __restrict__ coeffs_ws, float* __restrict__ outE)
{
  __shared__ float H1[S_][128];   // latent hidden (relu)
  __shared__ float LT[S_][64];    // latent
  __shared__ float Ysh[S_][16];   // SH basis at sphere samples
  __shared__ float BS[S_][3];     // basis MLP output

  int b = blockIdx.x, tid = threadIdx.x;
  const float* Fb = F + (size_t)b * S_ * 256;

  if (tid < S_) {
    float sx = S2[tid*3+0], sy = S2[tid*3+1], sz = S2[tid*3+2];
    float P[16]; solid_sh(sx, sy, sz, 1.0f, P);   // unit vectors
    for (int m = 0; m < 16; ++m) Ysh[tid][m] = P[m];
  }
  // latent hidden layer: [64,256]@[256,128] + relu
  for (int idx = tid; idx < S_*128; idx += 256) {
    int v = idx >> 7, o = idx & 127;
    float s = cb1[o];
    const float* fr = Fb + v*256;
    for (int k = 0; k < 256; ++k) s = fmaf(fr[k], cw1[k*128 + o], s);
    H1[v][o] = s > 0.f ? s : 0.f;
  }
  // basis branch fused: threads 0..63 each handle one sphere sample
  if (tid < S_) {
    const float* fr = Fb + tid*256;
    float s0 = bb2[0], s1 = bb2[1], s2 = bb2[2];
    for (int k = 0; k < 64; ++k) {
      float h = bb1[k];
      for (int j = 0; j < 256; ++j) h = fmaf(fr[j], bw1[j*64 + k], h);
      h = h > 0.f ? h : 0.f;
      s0 = fmaf(h, bw2[k*3+0], s0);
      s1 = fmaf(h, bw2[k*3+1], s1);
      s2 = fmaf(h, bw2[k*3+2], s2);
    }
    BS[tid][0] = s0; BS[tid][1] = s1; BS[tid][2] = s2;
  }
  __syncthreads();
  // latent layer 2: [64,128]@[128,64]
  for (int idx = tid; idx < S_*64; idx += 256) {
    int v = idx >> 6, o = idx & 63;
    float s = cb2[o];
    for (int k = 0; k < 128; ++k) s = fmaf(H1[v][k], cw2[k*64 + o], s);
    LT[v][o] = s;
  }
  __syncthreads();
  // SHT: coeffs[m][i] = quad * sum_v Ysh[v][m] * LT[v][i]
  for (int idx = tid; idx < 16*64; idx += 256) {
    int m = idx >> 6, i = idx & 63;
    float s = 0.f;
    for (int v = 0; v < S_; ++v) s = fmaf(Ysh[v][m], LT[v][i], s);
    coeffs_ws[(size_t)b*1024 + idx] = s * QUAD;
  }
  // E rows (m over [y,z,x] = SH indices 1..3), normalized along c
  if (tid < 3) {
    int m = tid;
    float e0 = 0.f, e1 = 0.f, e2 = 0.f;
    for (int v = 0; v < S_; ++v) {
      float y = Ysh[v][1 + m];
      e0 = fmaf(y, BS[v][0], e0);
      e1 = fmaf(y, BS[v][1], e1);
      e2 = fmaf(y, BS[v][2], e2);
    }
    e0 *= QUAD; e1 *= QUAD; e2 *= QUAD;
    float n = fmaxf(sqrtf(e0*e0 + e1*e1 + e2*e2), 1e-6f);
    outE[b*9 + m*3 + 0] = e0 / n;
    outE[b*9 + m*3 + 1] = e1 / n;
    outE[b*9 + m*3 + 2] = e2 / n;
  }
}

// ---------------------------------------------------------------------------
// Stage 2: per-point WMMA GEMM P[32,16] @ Cexp[16,256] -> q, radial expansion,
// points_code / coords / grid-sampled density.
// ---------------------------------------------------------------------------
__global__ __launch_bounds__(128) void stage2_kernel(
    const float* __restrict__ x, const float* __restrict__ dens,
    const float* __restrict__ coeffs, const float* __restrict__ iw,
    const float* __restrict__ ib, float* __restrict__ out)
{
  __shared__ float Cexp[256][18];   // transposed [n][m]; pad 18 -> aligned b64 B-frags
  __shared__ float Ps[128][18];     // solid harmonics per point (pad 18)
  __shared__ float r2s[128];
  __shared__ float Qw[4][32][66];   // per-wave D staging [point][qcol]
  __shared__ float iwS[384];
  __shared__ float ibS[3];

  int tid  = threadIdx.x;
  int lane = tid & 31, wave = tid >> 5;
  int half = lane >> 4, lan = lane & 15;
  int b    = blockIdx.x >> 8;       // 256 blocks per batch
  int tile = blockIdx.x & 255;

  // Build block-diagonal expanded coeff matrix (transposed): Cexp[n=l*64+i][m]
  const float* cb = coeffs + (size_t)b * 1024;
  for (int idx = tid; idx < 256*18; idx += 128) {
    int n = idx / 18, m = idx % 18;
    int l = n >> 6, mlo = l * l;
    float v = 0.f;
    if (m >= mlo && m < mlo + 2*l + 1) v = cb[m*64 + (n & 63)];
    Cexp[n][m] = v;
  }
  for (int idx = tid; idx < 384; idx += 128) iwS[idx] = iw[idx];
  if (tid < 3) ibS[tid] = ib[tid];

  // Per-point solid harmonics
  size_t pg = (size_t)b*V_ + (size_t)tile*128 + tid;
  float X = x[pg*3+0], Y = x[pg*3+1], Z = x[pg*3+2];
  float r2 = X*X + Y*Y + Z*Z;
  {
    float P[16]; solid_sh(X, Y, Z, r2, P);
    for (int m = 0; m < 16; ++m) Ps[tid][m] = P[m];
    Ps[tid][16] = 0.f; Ps[tid][17] = 0.f;
    r2s[tid] = r2;
  }
  __syncthreads();

  // A fragments: 16x4 f32, lane holds M=lane%16, VGPR {x,y} = K {2*half, 2*half+1}
  int pb = wave * 32;
  v2f a[2][4];
#pragma unroll
  for (int mt = 0; mt < 2; ++mt)
#pragma unroll
    for (int kk = 0; kk < 4; ++kk) {
      int row = pb + mt*16 + lan;
      int k   = 4*kk + 2*half;
      a[mt][kk].x = Ps[row][k];
      a[mt][kk].y = Ps[row][k+1];
    }

  float (*Qs)[66]     = Qw[wave];
  float* out_sampled  = out;
  float* out_coords   = out + OFF_COORDS;
  float* out_pc       = out + OFF_PC;
  size_t pgbase       = (size_t)b*V_ + (size_t)tile*128 + pb;

  for (int g = 0; g < 4; ++g) {          // N groups of 64 = SH degrees l=0..3
#pragma unroll
    for (int nt = 0; nt < 4; ++nt) {
      int n = g*64 + nt*16 + lan;
      v8f acc0 = {0.f,0.f,0.f,0.f,0.f,0.f,0.f,0.f};
      v8f acc1 = {0.f,0.f,0.f,0.f,0.f,0.f,0.f,0.f};
#pragma unroll
      for (int kk = 0; kk < 4; ++kk) {   // chain K=16 via 4x WMMA 16x16x4
        int k = 4*kk + 2*half;
        v2f bf; bf.x = Cexp[n][k]; bf.y = Cexp[n][k+1];
        acc0 = __builtin_amdgcn_wmma_f32_16x16x4_f32(false, a[0][kk], false, bf,
                                                     (short)0, acc0, false, false);
        acc1 = __builtin_amdgcn_wmma_f32_16x16x4_f32(false, a[1][kk], false, bf,
                                                     (short)0, acc1, false, false);
      }
      // D layout: element r -> point = mtile*16 + r + 8*half, col = nt*16 + lan
#pragma unroll
      for (int r = 0; r < 8; ++r) {
        Qs[r + 8*half][nt*16 + lan]      = acc0[r];
        Qs[16 + r + 8*half][nt*16 + lan] = acc1[r];
      }
    }
    __syncthreads();

    if (g < 2) {
      // l=0,1: 128 interleaved channels q[i]*{1, r^2} at offset g*128
      for (int p = 0; p < 32; ++p) {
        float rr = r2s[pb + p];
        float q0 = Qs[p][2*lane], q1 = Qs[p][2*lane + 1];
        float4 v4 = make_float4(q0, q0*rr, q1, q1*rr);
        size_t o = (pgbase + p)*384 + (size_t)(g*128 + 4*lane);
        *reinterpret_cast<float4*>(out_pc + o) = v4;
      }
    } else {
      // l=2,3: 64 direct channels at offset 256 + (g-2)*64
      for (int p = 0; p < 32; ++p) {
        float2 v2 = make_float2(Qs[p][2*lane], Qs[p][2*lane + 1]);
        size_t o = (pgbase + p)*384 + (size_t)(256 + (g-2)*64 + 2*lane);
        *reinterpret_cast<float2*>(out_pc + o) = v2;
      }
    }

    if (g == 1) {
      // points_inv @ iw + ib, coords, trilinear grid sample (lane = own point)
      float rr = r2s[pb + lane];
      float p0 = ibS[0], p1 = ibS[1], p2 = ibS[2];
      for (int i = 0; i < 64; ++i) {
        float q  = Qs[lane][i];
        float w0 = iwS[6*i+0] + rr*iwS[6*i+3];
        float w1 = iwS[6*i+1] + rr*iwS[6*i+4];
        float w2 = iwS[6*i+2] + rr*iwS[6*i+5];
        p0 = fmaf(q, w0, p0); p1 = fmaf(q, w1, p1); p2 = fmaf(q, w2, p2);
      }
      size_t pgl = pgbase + lane;
      out_coords[pgl*3+0] = p0;
      out_coords[pgl*3+1] = p1;
      out_coords[pgl*3+2] = p2;

      float ixf = (p0 + 1.f) * 15.5f;
      float iyf = (p1 + 1.f) * 15.5f;
      float izf = (p2 + 1.f) * 15.5f;
      float x0f = floorf(ixf), y0f = floorf(iyf), z0f = floorf(izf);
      float wx = ixf - x0f, wy = iyf - y0f, wz = izf - z0f;
      int x0 = (int)x0f, y0 = (int)y0f, z0 = (int)z0f;
      const float* volb = dens + (size_t)b * V_;
      auto tap = [&](int zi, int yi, int xi) -> float {
        bool in = ((unsigned)zi < 32u) && ((unsigned)yi < 32u) && ((unsigned)xi < 32u);
        int zc = zi < 0 ? 0 : (zi > 31 ? 31 : zi);
        int yc = yi < 0 ? 0 : (yi > 31 ? 31 : yi);
        int xc = xi < 0 ? 0 : (xi > 31 ? 31 : xi);
        float v = volb[(zc << 10) + (yc << 5) + xc];
        return in ? v : 0.f;
      };
      float s =
          tap(z0,   y0,   x0  ) * (1.f-wz)*(1.f-wy)*(1.f-wx)
        + tap(z0,   y0,   x0+1) * (1.f-wz)*(1.f-wy)*wx
        + tap(z0,   y0+1, x0  ) * (1.f-wz)*wy*(1.f-wx)
        + tap(z0,   y0+1, x0+1) * (1.f-wz)*wy*wx
        + tap(z0+1, y0,   x0  ) * wz*(1.f-wy)*(1.f-wx)
        + tap(z0+1, y0,   x0+1) * wz*(1.f-wy)*wx
        + tap(z0+1, y0+1, x0  ) * wz*wy*(1.f-wx)
        + tap(z0+1, y0+1, x0+1) * wz*wy*wx;
      out_sampled[pgl] = s;
    }
    __syncthreads();
  }
}

extern "C" void kernel_launch(void* const* d_in, const int* in_sizes, int n_in,
                              void* d_out, int out_size, void* d_ws, size_t ws_size,
                              hipStream_t stream) {
  const float* x    = (const float*)d_in[0];
  const float* dens = (const float*)d_in[1];
  const float* F    = (const float*)d_in[2];
  const float* S2   = (const float*)d_in[3];
  const float* bw1  = (const float*)d_in[4];
  const float* bb1  = (const float*)d_in[5];
  const float* bw2  = (const float*)d_in[6];
  const float* bb2  = (const float*)d_in[7];
  const float* cw1  = (const float*)d_in[8];
  const float* cb1  = (const float*)d_in[9];
  const float* cw2  = (const float*)d_in[10];
  const float* cb2  = (const float*)d_in[11];
  const float* iw   = (const float*)d_in[12];
  const float* ib   = (const float*)d_in[13];
  float* out       = (float*)d_out;
  float* coeffs_ws = (float*)d_ws;   // 8*16*64 floats = 32 KB

  stage1_kernel<<<B_, 256, 0, stream>>>(F, S2, bw1, bb1, bw2, bb2,
                                        cw1, cb1, cw2, cb2,
                                        coeffs_ws, out + OFF_E);
  stage2_kernel<<<2048, 128, 0, stream>>>(x, dens, coeffs_ws, iw, ib, out);
}